// BigramBaseline_17506286698648
// MI455X (gfx1250) — compile-verified
//
#include <hip/hip_runtime.h>

// Problem constants (from reference): idx [8,2048] int, emb [4096,4096] f32,
// out = causal prefix-mean of gathered rows, [8*2048, 4096] f32.
#define VOCAB 4096
#define BATCH 8
#define TLEN  2048

#define CHUNK 128            // timesteps handled by one kernel-2 workgroup
#define NC    (TLEN / CHUNK) // 16 chunks along T
#define SUB   16             // WMMA tile: 16 timesteps x 16 columns
#define NSUB  (CHUNK / SUB)  // 8 sequential WMMA sub-scans per workgroup
#define COLS  64             // columns per workgroup -> 4 wave32s
#define WAVES (COLS / 16)    // 4 waves, one 16-col WMMA tile each
#define LDS_STRIDE 80        // 64 + 16 pad: row k vs k+1 hit disjoint bank groups
#define K1_THREADS 256

typedef __attribute__((ext_vector_type(2))) float v2f;
typedef __attribute__((ext_vector_type(4))) float v4f;
typedef __attribute__((ext_vector_type(8))) float v8f;

// ---------------------------------------------------------------------------
// Kernel 1: per-chunk column sums S[b, chunk, col] = sum over 128 gathered rows.
// Fully coalesced: 256 consecutive columns per block = 1 KB contiguous per row.
// ---------------------------------------------------------------------------
__global__ void __launch_bounds__(K1_THREADS)
chunk_sums_kernel(const int* __restrict__ idx,
                  const float* __restrict__ emb,
                  float* __restrict__ S) {
    __shared__ int rows[CHUNK];
    const int bc  = blockIdx.x;            // b * NC + chunk
    const int b   = bc / NC;
    const int c   = bc % NC;
    const int tid = threadIdx.x;

    if (tid < CHUNK) rows[tid] = idx[b * TLEN + c * CHUNK + tid];
    __syncthreads();

    const int col = blockIdx.y * K1_THREADS + tid;
    float acc = 0.0f;
#pragma unroll 8
    for (int r = 0; r < CHUNK; ++r)
        acc += emb[(long)rows[r] * VOCAB + col];
    S[(long)bc * VOCAB + col] = acc;
}

// ---------------------------------------------------------------------------
// Kernel 2: per (batch, chunk, 64-col block):
//   1) stage 128x64 gather tile into LDS (coalesced b128 loads); also compute
//      the 128 reciprocals 1/(t+1) once per block into LDS (off hot path)
//   2) carry = sum of preceding chunks' column sums (from workspace)
//   3) 8x sequential 16-step sub-scans: D = L16x16 * X + carry via
//      V_WMMA_F32_16X16X4_F32 (4 chained K=4 slices); carry forwarded from
//      D row 15 with one ds_bpermute broadcast.
//   4) scale by precomputed 1/(t+1), store non-temporally (output is
//      write-once; keep emb/S resident in the 192 MB L2) through a single
//      per-lane base pointer + immediate offsets.
// ---------------------------------------------------------------------------
__global__ void __launch_bounds__(WAVES * 32)
prefix_mean_wmma_kernel(const int* __restrict__ idx,
                        const float* __restrict__ emb,
                        const float* __restrict__ S,
                        float* __restrict__ out) {
    __shared__ float xtile[CHUNK * LDS_STRIDE]; // 40 KB
    __shared__ float rcp_lds[CHUNK];            // 1/(t+1) per chunk row
    __shared__ int   rows[CHUNK];

    const int cb   = blockIdx.x;   // column block: 0 .. VOCAB/COLS-1
    const int c    = blockIdx.y;   // chunk index
    const int b    = blockIdx.z;   // batch
    const int tid  = threadIdx.x;  // 0..127
    const int lane = tid & 31;
    const int wave = tid >> 5;
    const int half = lane >> 4;    // 0: lanes 0-15, 1: lanes 16-31
    const int l16  = lane & 15;

    if (tid < CHUNK) {
        rows[tid]    = idx[b * TLEN + c * CHUNK + tid];
        // full-precision divide, once per block -> off the hot path
        rcp_lds[tid] = 1.0f / (float)(c * CHUNK + tid + 1);
    }
    __syncthreads();

    // --- stage gather tile: wave handles 2 rows/iter (256 B contiguous per
    // half-wave), 4 waves * 16 iters * 2 rows = 128 rows.
    const int colbase = cb * COLS;
#pragma unroll 4
    for (int it = 0; it < CHUNK / (WAVES * 2); ++it) {
        const int r = (it * WAVES + wave) * 2 + half;
        const float* src = emb + (long)rows[r] * VOCAB + colbase + l16 * 4;
        v4f v = *(const v4f*)src;                       // global_load_b128
        *(v4f*)&xtile[r * LDS_STRIDE + l16 * 4] = v;    // ds_store_b128, no conflicts
    }
    __syncthreads();

    // --- carry from preceding chunks (decoupled scan over workspace sums)
    const int mycol = colbase + wave * 16 + l16;
    float carry = 0.0f;
    for (int cc = 0; cc < c; ++cc)
        carry += S[(long)(b * NC + cc) * VOCAB + mycol];

    // --- A-matrix slices of the unit lower-triangular L (constant per lane).
    // A 16x4 f32 layout: all lanes row M = lane&15; VGPR0 holds K = half*2,
    // VGPR1 holds K = half*2 + 1.
    const int kb = half * 2;
    v2f A[4];
#pragma unroll
    for (int j = 0; j < 4; ++j) {
        A[j].x = (4 * j + kb     <= l16) ? 1.0f : 0.0f;
        A[j].y = (4 * j + kb + 1 <= l16) ? 1.0f : 0.0f;
    }

    // --- per-lane output base: fold batch/chunk/half/column in once; the 8
    // rows of a sub-block are then fixed immediate offsets r*VOCAB, and each
    // sub-block advances the base by SUB*VOCAB (one 64-bit add / iteration).
    float* outp = out + ((long)(b * TLEN + c * CHUNK + half * 8) * VOCAB + mycol);
    const int rbase = half * 8;    // this lane's first D row within sub-block

    // --- sequential 16-step sub-scans
    const float* ldsc = &xtile[wave * 16 + l16];  // this lane's column in LDS
#pragma unroll 1
    for (int s = 0; s < NSUB; ++s) {
        v8f Cacc;
#pragma unroll
        for (int r = 0; r < 8; ++r) Cacc[r] = carry;

        const int rowbase = s * SUB;
        // B 4x16 f32 layout per K-slice j: VGPR0 = rows {4j, 4j+1} (half 0/1),
        // VGPR1 = rows {4j+2, 4j+3}. LDS stride 80 => the two rows map to
        // disjoint 16-bank groups.
#pragma unroll
        for (int j = 0; j < 4; ++j) {
            v2f Bv;
            Bv.x = ldsc[(rowbase + 4 * j + half)     * LDS_STRIDE];
            Bv.y = ldsc[(rowbase + 4 * j + 2 + half) * LDS_STRIDE];
            Cacc = __builtin_amdgcn_wmma_f32_16x16x4_f32(
                false, A[j], false, Bv, (short)0, Cacc, false, false);
        }

        // scale by precomputed 1/(t+1); streaming (non-temporal) stores,
        // 64 B contiguous per 16-lane group, immediate-offset addressing.
#pragma unroll
        for (int r = 0; r < 8; ++r)
            __builtin_nontemporal_store(Cacc[r] * rcp_lds[rowbase + rbase + r],
                                        outp + r * VOCAB);
        outp += SUB * VOCAB;

        // next carry = D row 15 (= old carry + column sum of this sub-block),
        // held in VGPR7 of lanes 16..31; broadcast to both halves.
        carry = __shfl(Cacc[7], 16 + l16, 32);
    }
}

// ---------------------------------------------------------------------------
extern "C" void kernel_launch(void* const* d_in, const int* in_sizes, int n_in,
                              void* d_out, int out_size, void* d_ws, size_t ws_size,
                              hipStream_t stream) {
    const int*   idx = (const int*)d_in[0];
    const float* emb = (const float*)d_in[1];
    float*       out = (float*)d_out;
    float*       S   = (float*)d_ws;   // B*NC*VOCAB floats = 2 MB chunk sums

    dim3 g1(BATCH * NC, VOCAB / K1_THREADS);           // 128 x 16 blocks
    chunk_sums_kernel<<<g1, K1_THREADS, 0, stream>>>(idx, emb, S);

    dim3 g2(VOCAB / COLS, NC, BATCH);                  // 64 x 16 x 8 = 8192 blocks
    prefix_mean_wmma_kernel<<<g2, WAVES * 32, 0, stream>>>(idx, emb, S, out);
}